// GNN_24249385353613
// MI455X (gfx1250) — compile-verified
//
#include <hip/hip_runtime.h>

#define NN 50000
#define NE 800000
#define D  128
#define MTILES (NN / 16)   // 3125 exactly
#define NTILES (D / 16)    // 8

typedef __attribute__((ext_vector_type(2))) float v2f;
typedef __attribute__((ext_vector_type(8))) float v8f;

// ---------------------------------------------------------------------------
// Kernel 1: zero the scatter workspace (sum rows + degree counts)
// ---------------------------------------------------------------------------
__global__ __launch_bounds__(256) void sage_zero(float* __restrict__ p, int n) {
  int i = blockIdx.x * blockDim.x + threadIdx.x;
  if (i < n) p[i] = 0.0f;
}

// ---------------------------------------------------------------------------
// Kernel 2: edge scatter. 32 lanes per edge; each lane moves 4 contiguous
// features (float4 gather from x[src], 4 HW f32 atomics into ws_sum[dst]).
// x (25.6MB) and ws_sum (25.6MB) both live in the 192MB L2, so this runs at
// L2/atomic-unit rate. Lane 0 accumulates the degree count as float.
// ---------------------------------------------------------------------------
__global__ __launch_bounds__(256) void sage_scatter(
    const float* __restrict__ x,
    const long long* __restrict__ src,
    const long long* __restrict__ dst,
    float* __restrict__ ws_sum,
    float* __restrict__ ws_cnt) {
  unsigned int gtid = blockIdx.x * 256u + threadIdx.x;
  unsigned int edge = gtid >> 5;          // NE*32 threads exactly
  unsigned int lane = gtid & 31u;

  int s = (int)src[edge];
  int d = (int)dst[edge];

  const float4 v = *(const float4*)(x + (size_t)s * D + lane * 4);
  float* p = ws_sum + (size_t)d * D + lane * 4;
  unsafeAtomicAdd(p + 0, v.x);            // global_atomic_add_f32 (no return)
  unsafeAtomicAdd(p + 1, v.y);
  unsafeAtomicAdd(p + 2, v.z);
  unsafeAtomicAdd(p + 3, v.w);
  if (lane == 0) unsafeAtomicAdd(ws_cnt + d, 1.0f);
}

// ---------------------------------------------------------------------------
// Kernel 3: fused mean + dual GEMM + bias using V_WMMA_F32_16X16X4_F32.
//   out[m,n] = sum_k (ws_sum[m,k]/max(cnt[m],1))*Wl[n,k] + x[m,k]*Wr[n,k] + b[n]
// One wave per 16x16 output tile. Block = 8 waves = one M-tile across all 8
// N-tiles (A fragments shared through L0/L1).
//
// Fragment layouts per CDNA5 ISA (fp32, wave32):
//   A 16x4 : lane L holds row M=L&15, K = k0 + 2*(L>>4) + {0,1} in 2 VGPRs
//   B 4x16 : lane L holds col N=L&15, K = k0 + 2*(L>>4) + {0,1} in 2 VGPRs
//   C 16x16: 8 VGPRs; lane L col N=L&15, VGPR v -> row M = v + 8*(L>>4)
// ---------------------------------------------------------------------------
__global__ __launch_bounds__(256) void sage_gemm(
    const float* __restrict__ x,
    const float* __restrict__ Wl,
    const float* __restrict__ Wr,
    const float* __restrict__ bias,
    const float* __restrict__ ws_sum,
    const float* __restrict__ ws_cnt,
    float* __restrict__ out) {
  const int lane   = threadIdx.x & 31;
  const int m_tile = blockIdx.x;            // 0..3124
  const int n_tile = threadIdx.x >> 5;      // 0..7

  const int mn   = lane & 15;               // sub-row for A / sub-col for B,C,D
  const int koff = (lane >> 4) * 2;         // K phase for A/B fragments

  const int m = m_tile * 16 + mn;           // node row this lane feeds into A
  const int n = n_tile * 16 + mn;           // output column this lane feeds into B

  const float inv = 1.0f / fmaxf(ws_cnt[m], 1.0f);   // mean scale, folded into A

  const float* __restrict__ arow_s = ws_sum + (size_t)m * D;
  const float* __restrict__ arow_x = x      + (size_t)m * D;
  const float* __restrict__ wl_row = Wl     + (size_t)n * D;   // B[k,n] = W[n,k]
  const float* __restrict__ wr_row = Wr     + (size_t)n * D;

  v8f acc = {};

#pragma unroll
  for (int k0 = 0; k0 < D; k0 += 4) {
    const int kb = k0 + koff;               // 8B-aligned (k0%4==0, koff in {0,2})

    // agg @ Wl^T  (mean folded into the A fragment)
    v2f a1 = *(const v2f*)(arow_s + kb);
    a1 = a1 * inv;
    v2f b1 = *(const v2f*)(wl_row + kb);
    acc = __builtin_amdgcn_wmma_f32_16x16x4_f32(
        /*neg_a=*/false, a1, /*neg_b=*/false, b1,
        /*c_mod=*/(short)0, acc, /*reuse_a=*/false, /*reuse_b=*/false);

    // x @ Wr^T
    v2f a2 = *(const v2f*)(arow_x + kb);
    v2f b2 = *(const v2f*)(wr_row + kb);
    acc = __builtin_amdgcn_wmma_f32_16x16x4_f32(
        false, a2, false, b2, (short)0, acc, false, false);
  }

  // Bias + store. C/D layout: lane col = mn, VGPR v -> row v + 8*(lane>>4).
  const float bv   = bias[n_tile * 16 + mn];
  const int   col  = n_tile * 16 + mn;
  const int   rhi  = (lane >> 4) * 8;
  float* __restrict__ obase = out + (size_t)(m_tile * 16 + rhi) * D + col;
#pragma unroll
  for (int v = 0; v < 8; ++v) {
    obase[(size_t)v * D] = acc[v] + bv;
  }
}

// ---------------------------------------------------------------------------
// Host-side launcher
// ---------------------------------------------------------------------------
extern "C" void kernel_launch(void* const* d_in, const int* in_sizes, int n_in,
                              void* d_out, int out_size, void* d_ws, size_t ws_size,
                              hipStream_t stream) {
  const float*     x  = (const float*)d_in[0];
  const long long* ei = (const long long*)d_in[1];   // int64 edge_index [2, NE]
  const float*     Wl = (const float*)d_in[2];
  const float*     Wr = (const float*)d_in[3];
  const float*     b  = (const float*)d_in[4];
  float*           out = (float*)d_out;

  float* ws_sum = (float*)d_ws;                      // [NN, D]
  float* ws_cnt = ws_sum + (size_t)NN * D;           // [NN]

  const int zn = NN * D + NN;
  sage_zero<<<(zn + 255) / 256, 256, 0, stream>>>(ws_sum, zn);

  // NE edges * 32 lanes = 25.6M threads = exactly 100000 blocks of 256
  sage_scatter<<<(NE * 32) / 256, 256, 0, stream>>>(x, ei, ei + NE, ws_sum, ws_cnt);

  // One block (8 waves) per 16-node M tile; wave w handles N-tile w.
  sage_gemm<<<MTILES, 256, 0, stream>>>(x, Wl, Wr, b, ws_sum, ws_cnt, out);
}